// I24EMA_19490561590092
// MI455X (gfx1250) — compile-verified
//
#include <hip/hip_runtime.h>

// EMA scan y_t = a*y_{t-1} + (1-a)*x_t over (B=8, T=16384, C=256) f32.
// Strategy: 16-step chunks computed as D = M*X + carry-injection via
// 4 chained V_WMMA_F32_16X16X4_F32 (full-precision f32 WMMA).
// Time parallelism via 256-step lookback windows (0.9^256 ~ 2e-12 << f32 ulp),
// giving 2048 independent wave32 blocks -> enough to saturate 23.3 TB/s HBM.

typedef __attribute__((ext_vector_type(2))) float v2f;
typedef __attribute__((ext_vector_type(8))) float v8f;

#define EMA_ALPHA 0.9f
#define EMA_OMA   0.1f   // 1 - alpha

namespace {
constexpr int Bn  = 8;
constexpr int Tn  = 16384;
constexpr int Cn  = 256;
constexpr int SEG = 1024;   // output timesteps per block
constexpr int WIN = 256;    // lookback warm-up (alpha^256 ~ 1.9e-12)
constexpr int CH  = 16;     // channels per tile (WMMA N)
}

__global__ __launch_bounds__(32)
void ema_wmma_kernel(const float* __restrict__ x, float* __restrict__ y)
{
    // alpha^k, k = 0..16 (compile-time, only constant-indexed)
    constexpr float P[17] = {
        1.0f, 0.9f, 0.81f, 0.729f, 0.6561f, 0.59049f, 0.531441f,
        0.4782969f, 0.43046721f, 0.387420489f, 0.3486784401f,
        0.31381059609f, 0.282429536481f, 0.2541865828329f,
        0.22876792454961f, 0.205891132094649f, 0.1853020188851841f};
    // alpha^-j, j = 0..15
    constexpr float Pi[16] = {
        1.0f, 1.11111111f, 1.23456790f, 1.37174211f, 1.52415790f,
        1.69350878f, 1.88167642f, 2.09075158f, 2.32305731f, 2.58117479f,
        2.86797199f, 3.18663554f, 3.54070615f, 3.93411795f, 4.37124216f,
        4.85693573f};

    const int lane = (int)threadIdx.x;   // 0..31
    const int n    = lane & 15;          // channel within tile / A-frag row M
    const int kh   = lane >> 4;          // which K-half this lane holds

    const int seg = (int)blockIdx.x;     // time segment
    const int ct  = (int)blockIdx.y;     // channel tile
    const int b   = (int)blockIdx.z;     // batch

    const int t_out0  = seg * SEG;
    const int t_begin = (seg == 0) ? 0 : (t_out0 - WIN);
    const int nchunk  = (t_out0 + SEG - t_begin) >> 4;
    const int skip    = (t_out0 - t_begin) >> 4;   // warm-up chunks (no store)

    // pm = alpha^m for this lane's A-matrix row m = n
    float pm = 1.0f;
    #pragma unroll
    for (int k = 0; k < 15; ++k) pm = (k < n) ? pm * EMA_ALPHA : pm;

    // A fragments for the 4 chained K=4 WMMAs.
    // A layout (f32 16x4): lane = (K>=2 ? 16:0) + M, component = K&1.
    // coef(m, j) = (m>=j) ? (1-a)*a^(m-j) : 0,  j = 4q + 2*kh + comp
    v2f a[4];
    #pragma unroll
    for (int q = 0; q < 4; ++q) {
        #pragma unroll
        for (int comp = 0; comp < 2; ++comp) {
            const int   base = 4 * q + comp;
            const float inv  = kh ? Pi[base + 2] : Pi[base];
            const int   j    = base + 2 * kh;
            a[q][comp] = (n >= j) ? (EMA_OMA * pm) * inv : 0.0f;
        }
    }

    // carry-injection multipliers alpha^{row+1}, row = v + 8*kh (C/D layout)
    float cmul[8];
    #pragma unroll
    for (int v = 0; v < 8; ++v) cmul[v] = kh ? P[v + 9] : P[v + 1];

    const size_t baseElem = ((size_t)b * Tn + (size_t)t_begin) * Cn
                          + (size_t)ct * CH + (size_t)n;
    const float* xp = x + baseElem;
    float*       yp = y + baseElem;

    float carry = 0.0f;

    // Software pipeline: preload chunk 0.
    // B layout (f32 4x16): lane half selects K-pair, column N = lane&15.
    float cur[8], nxt[8];
    #pragma unroll
    for (int e = 0; e < 8; ++e) nxt[e] = 0.0f;
    #pragma unroll
    for (int q = 0; q < 4; ++q) {
        const int r = 4 * q + 2 * kh;
        cur[2 * q]     = xp[(size_t)r * Cn];
        cur[2 * q + 1] = xp[(size_t)(r + 1) * Cn];
    }

    for (int i = 0; i < nchunk; ++i) {
        const float* xc_next = xp + (size_t)(i + 1) * 16 * Cn;
        if (i + 2 < nchunk)
            __builtin_prefetch(xp + (size_t)(i + 2) * 16 * Cn, 0, 1);
        if (i + 1 < nchunk) {
            #pragma unroll
            for (int q = 0; q < 4; ++q) {
                const int r = 4 * q + 2 * kh;
                nxt[2 * q]     = xc_next[(size_t)r * Cn];
                nxt[2 * q + 1] = xc_next[(size_t)(r + 1) * Cn];
            }
        }

        // acc init: alpha^{row+1} * carry
        v8f acc;
        #pragma unroll
        for (int v = 0; v < 8; ++v) acc[v] = cmul[v] * carry;

        // D = M * X + acc  via 4 chained K=4 f32 WMMAs
        #pragma unroll
        for (int q = 0; q < 4; ++q) {
            v2f bq;
            bq[0] = cur[2 * q];
            bq[1] = cur[2 * q + 1];
            acc = __builtin_amdgcn_wmma_f32_16x16x4_f32(
                /*neg_a=*/false, a[q], /*neg_b=*/false, bq,
                /*c_mod=*/(short)0, acc, /*reuse_a=*/false, /*reuse_b=*/false);
        }

        if (i >= skip) {
            float* yc = yp + (size_t)i * 16 * Cn;
            #pragma unroll
            for (int v = 0; v < 8; ++v) {
                const int row = v + 8 * kh;
                yc[(size_t)row * Cn] = acc[v];
            }
        }

        // carry = D[15][n] : acc[7] held by lane 16+n (uniform per column n)
        carry = __int_as_float(__builtin_amdgcn_ds_bpermute(
            (16 + n) << 2, __float_as_int(acc[7])));

        #pragma unroll
        for (int e = 0; e < 8; ++e) cur[e] = nxt[e];
    }
}

extern "C" void kernel_launch(void* const* d_in, const int* in_sizes, int n_in,
                              void* d_out, int out_size, void* d_ws, size_t ws_size,
                              hipStream_t stream) {
    (void)in_sizes; (void)n_in; (void)out_size; (void)d_ws; (void)ws_size;
    const float* x = (const float*)d_in[0];
    float*       y = (float*)d_out;
    dim3 grid(Tn / SEG, Cn / CH, Bn);   // (16, 16, 8) = 2048 wave32 blocks
    ema_wmma_kernel<<<grid, 32, 0, stream>>>(x, y);
}